// ComputeAllAtomCoords_3307124817981
// MI455X (gfx1250) — compile-verified
//
#include <hip/hip_runtime.h>
#include <hip/hip_bf16.h>
#include <cstdint>

// ComputeAllAtomCoords (RoseTTAFold-style) for MI455X / gfx1250.
// One residue per lane (wave32). Per-AA constant tables staged into LDS via
// the Tensor Data Mover (tensor_load_to_lds + s_wait_tensorcnt); main math is
// per-thread fp32 (bandwidth-bound op: ~67 MB traffic, ~0.3 GFLOP).

namespace {

constexpr int kNAA    = 22;
constexpr int kNATOMS = 27;
constexpr int kNRT    = 7;                    // RT frames per AA in table
constexpr int kRTF    = kNAA * kNRT * 16;     // 2464 f32
constexpr int kBXF    = kNAA * kNATOMS * 4;   // 2376 f32
constexpr int kBII    = kNAA * kNATOMS;       //  594 i32

typedef unsigned int u32x4 __attribute__((ext_vector_type(4)));
typedef int          i32x4 __attribute__((ext_vector_type(4)));
typedef int          i32x8 __attribute__((ext_vector_type(8)));

#if defined(__HIP_DEVICE_COMPILE__) && defined(__gfx1250__) && \
    __has_builtin(__builtin_amdgcn_tensor_load_to_lds) && \
    __has_builtin(__builtin_amdgcn_s_wait_tensorcnt)
#define ATHENA_USE_TDM 1
#else
#define ATHENA_USE_TDM 0
#endif

#if ATHENA_USE_TDM
// Issue one TDM descriptor copying `ndw` dwords from global to LDS.
// D# layout per CDNA5 ISA §8 (groups 0/1; groups 2/3 zero => <=2D tensor).
__device__ __forceinline__ void tdm_copy_dwords_to_lds(const void* gsrc,
                                                       void* lds_dst,
                                                       unsigned ndw) {
  unsigned long long ga = (unsigned long long)(uintptr_t)gsrc;
  // Generic pointers to LDS carry the LDS byte offset in the low 32 bits.
  unsigned lds = (unsigned)(uintptr_t)lds_dst;

  u32x4 g0;
  g0.x = 1u;                                            // count=1 (valid), user mode
  g0.y = lds;                                           // lds_addr
  g0.z = (unsigned)(ga & 0xffffffffull);                // global_addr[31:0]
  g0.w = (unsigned)((ga >> 32) & 0x01ffffffull)         // global_addr[56:32]
       | (2u << 30);                                    // type=2 ("image")

  i32x8 g1;
  g1[0] = (int)(2u << 16);                              // data_size=2 -> 4 bytes
  g1[1] = (int)((ndw & 0xffffu) << 16);                 // tensor_dim0[15:0]
  g1[2] = (int)((ndw >> 16) | (1u << 16));              // tensor_dim0[31:16] | tensor_dim1=1
  g1[3] = (int)(ndw << 16);                             // tensor_dim1 hi=0 | tile_dim0=ndw
  g1[4] = 0;                                            // tile_dim1=0, tile_dim2=0 (unused)
  g1[5] = (int)ndw;                                     // tensor_dim0_stride (lo32)
  g1[6] = 0;                                            // stride0 hi | stride1 lo
  g1[7] = 0;                                            // stride1 hi

  i32x4 z4 = {0, 0, 0, 0};
#if __clang_major__ >= 23
  i32x8 z8 = {0, 0, 0, 0, 0, 0, 0, 0};
  __builtin_amdgcn_tensor_load_to_lds(g0, g1, z4, z4, z8, 0);
#else
  __builtin_amdgcn_tensor_load_to_lds(g0, g1, z4, z4, 0);
#endif
}
#endif  // ATHENA_USE_TDM

__device__ __forceinline__ void csn(float a0, float a1, float eps,
                                    float& c, float& s) {
  float n = sqrtf(fmaf(a0, a0, a1 * a1)) + eps;
  float r = 1.0f / n;
  c = a0 * r;
  s = a1 * r;
}

// D = A(4x4) * B(4x4), row-major. B may live in LDS.
__device__ __forceinline__ void mm44(float* __restrict__ D,
                                     const float* __restrict__ A,
                                     const float* __restrict__ B) {
#pragma unroll
  for (int i = 0; i < 4; ++i) {
    float a0 = A[4 * i + 0], a1 = A[4 * i + 1];
    float a2 = A[4 * i + 2], a3 = A[4 * i + 3];
#pragma unroll
    for (int j = 0; j < 4; ++j) {
      D[4 * i + j] =
          fmaf(a0, B[j], fmaf(a1, B[4 + j], fmaf(a2, B[8 + j], a3 * B[12 + j])));
    }
  }
}

// M = M @ rotX(c,s): col1' = c*col1 + s*col2 ; col2' = -s*col1 + c*col2
__device__ __forceinline__ void postRotX(float* M, float c, float s) {
#pragma unroll
  for (int i = 0; i < 4; ++i) {
    float m1 = M[4 * i + 1], m2 = M[4 * i + 2];
    M[4 * i + 1] = fmaf(c, m1, s * m2);
    M[4 * i + 2] = fmaf(c, m2, -s * m1);
  }
}

// M = M @ rotZ(c,s): col0' = c*col0 + s*col1 ; col1' = -s*col0 + c*col1
__device__ __forceinline__ void postRotZ(float* M, float c, float s) {
#pragma unroll
  for (int i = 0; i < 4; ++i) {
    float m0 = M[4 * i + 0], m1 = M[4 * i + 1];
    M[4 * i + 0] = fmaf(c, m0, s * m1);
    M[4 * i + 1] = fmaf(c, m1, -s * m0);
  }
}

// Rodrigues rotation about unit axis u with (cos,sin) from normalized angs.
__device__ __forceinline__ void rotAxis(float* A, float a0, float a1,
                                        float ux, float uy, float uz) {
  float ct, st;
  csn(a0, a1, 1e-6f, ct, st);
  float o = 1.0f - ct;
  A[0]  = fmaf(ux * ux, o, ct);
  A[1]  = fmaf(ux * uy, o, -uz * st);
  A[2]  = fmaf(ux * uz, o,  uy * st);
  A[3]  = 0.0f;
  A[4]  = fmaf(ux * uy, o,  uz * st);
  A[5]  = fmaf(uy * uy, o, ct);
  A[6]  = fmaf(uy * uz, o, -ux * st);
  A[7]  = 0.0f;
  A[8]  = fmaf(ux * uz, o, -uy * st);
  A[9]  = fmaf(uy * uz, o,  ux * st);
  A[10] = fmaf(uz * uz, o, ct);
  A[11] = 0.0f;
  A[12] = 0.0f; A[13] = 0.0f; A[14] = 0.0f; A[15] = 1.0f;
}

__device__ __forceinline__ void store_mat(float* dst, const float* M) {
  float4* d = reinterpret_cast<float4*>(dst);  // 16B aligned (l*576 + k*64)
  d[0] = make_float4(M[0],  M[1],  M[2],  M[3]);
  d[1] = make_float4(M[4],  M[5],  M[6],  M[7]);
  d[2] = make_float4(M[8],  M[9],  M[10], M[11]);
  d[3] = make_float4(M[12], M[13], M[14], M[15]);
}

}  // namespace

__global__ __launch_bounds__(256)
void allatom_kernel(const float* __restrict__ xyz,     // (L,3,3)
                    const float* __restrict__ alphas,  // (L,10,2)
                    const float* __restrict__ gRT,     // (22,7,4,4)
                    const float* __restrict__ gBX,     // (22,27,4)
                    const int*   __restrict__ seq,     // (L)
                    const int*   __restrict__ gBI,     // (22,27)
                    float* __restrict__ out_frames,    // (L,9,4,4)
                    float* __restrict__ out_xyz,       // (L,27,3)
                    int L) {
  __shared__ alignas(16) float sRT[kRTF];
  __shared__ alignas(16) float sBX[kBXF];
  __shared__ alignas(16) int   sBI[kBII];

#if ATHENA_USE_TDM
  if (threadIdx.x < 32) {  // wave 0 issues three async TDM descriptors
    tdm_copy_dwords_to_lds(gRT, sRT, (unsigned)kRTF);
    tdm_copy_dwords_to_lds(gBX, sBX, (unsigned)kBXF);
    tdm_copy_dwords_to_lds(gBI, sBI, (unsigned)kBII);
    __builtin_amdgcn_s_wait_tensorcnt(0);
  }
#else
  for (int i = threadIdx.x; i < kRTF; i += blockDim.x) sRT[i] = gRT[i];
  for (int i = threadIdx.x; i < kBXF; i += blockDim.x) sBX[i] = gBX[i];
  for (int i = threadIdx.x; i < kBII; i += blockDim.x) sBI[i] = gBI[i];
#endif
  __syncthreads();

  const int l = blockIdx.x * blockDim.x + threadIdx.x;
  if (l >= L) return;

  // ---------------- backbone frame RTF0 (rigid_from_3_points) --------------
  const float* p = xyz + (size_t)l * 9;
  float Nx = p[0], Ny = p[1], Nz = p[2];
  float Ax = p[3], Ay = p[4], Az = p[5];   // Ca
  float Cx = p[6], Cy = p[7], Cz = p[8];

  float v1x = Cx - Ax, v1y = Cy - Ay, v1z = Cz - Az;
  float v2x = Nx - Ax, v2y = Ny - Ay, v2z = Nz - Az;
  float r1 = 1.0f / (sqrtf(fmaf(v1x, v1x, fmaf(v1y, v1y, v1z * v1z))) + 1e-8f);
  float e1x = v1x * r1, e1y = v1y * r1, e1z = v1z * r1;
  float d = fmaf(e1x, v2x, fmaf(e1y, v2y, e1z * v2z));
  float u2x = fmaf(-d, e1x, v2x), u2y = fmaf(-d, e1y, v2y), u2z = fmaf(-d, e1z, v2z);
  float r2 = 1.0f / (sqrtf(fmaf(u2x, u2x, fmaf(u2y, u2y, u2z * u2z))) + 1e-8f);
  float e2x = u2x * r2, e2y = u2y * r2, e2z = u2z * r2;
  float e3x = e1y * e2z - e1z * e2y;
  float e3y = e1z * e2x - e1x * e2z;
  float e3z = e1x * e2y - e1y * e2x;

  float F0[16] = {e1x, e2x, e3x, Ax,
                  e1y, e2y, e3y, Ay,
                  e1z, e2z, e3z, Az,
                  0.f, 0.f, 0.f, 1.f};

  const int s = seq[l];
  const float* RTl = sRT + s * (kNRT * 16);
  const float* BXl = sBX + s * (kNATOMS * 4);
  const int*   BIl = sBI + s * kNATOMS;

  // alphas: 10 (cos,sin) pairs, 80B stride -> 5 aligned float4 loads
  const float4* a4 = reinterpret_cast<const float4*>(alphas + (size_t)l * 20);
  float4 q0 = a4[0], q1 = a4[1], q2 = a4[2], q3 = a4[3], q4 = a4[4];

  float* fbase = out_frames + (size_t)l * 144;
  store_mat(fbase + 0 * 16, F0);

  float c, sn;
  float T[16], M2[16];

  // RTF1..RTF3 = F0 @ RT[k] @ rotX(alpha[k])
  mm44(T, F0, RTl + 0 * 16); csn(q0.x, q0.y, 1e-6f, c, sn); postRotX(T, c, sn);
  store_mat(fbase + 1 * 16, T);
  mm44(T, F0, RTl + 1 * 16); csn(q0.z, q0.w, 1e-6f, c, sn); postRotX(T, c, sn);
  store_mat(fbase + 2 * 16, T);
  mm44(T, F0, RTl + 2 * 16); csn(q1.x, q1.y, 1e-6f, c, sn); postRotX(T, c, sn);
  store_mat(fbase + 3 * 16, T);

  // ---------------- CB rotation axes from base coordinates -----------------
  float N0x = BXl[0],  N0y = BXl[1],  N0z = BXl[2];
  float CAx = BXl[4],  CAy = BXl[5],  CAz = BXl[6];
  float C0x = BXl[8],  C0y = BXl[9],  C0z = BXl[10];
  float CBx = BXl[16], CBy = BXl[17], CBz = BXl[18];

  float NCrx = 0.5f * (C0x + N0x), NCry = 0.5f * (C0y + N0y), NCrz = 0.5f * (C0z + N0z);
  float vbx = CBx - CAx, vby = CBy - CAy, vbz = CBz - CAz;
  float vnx = NCrx - CAx, vny = NCry - CAy, vnz = NCrz - CAz;

  float a1x = vby * vnz - vbz * vny;
  float a1y = vbz * vnx - vbx * vnz;
  float a1z = vbx * vny - vby * vnx;
  float ra1 = 1.0f / (sqrtf(fmaf(a1x, a1x, fmaf(a1y, a1y, a1z * a1z))) + 1e-8f);
  a1x *= ra1; a1y *= ra1; a1z *= ra1;

  float NCpx = C0x - N0x, NCpy = C0y - N0y, NCpz = C0z - N0z;
  float t = fmaf(NCpx, NCrx, fmaf(NCpy, NCry, NCpz * NCrz)) /
            fmaf(NCrx, NCrx, fmaf(NCry, NCry, NCrz * NCrz));
  float ppx = fmaf(-t, NCrx, NCpx), ppy = fmaf(-t, NCry, NCpy), ppz = fmaf(-t, NCrz, NCpz);

  float a2x = vby * ppz - vbz * ppy;
  float a2y = vbz * ppx - vbx * ppz;
  float a2z = vbx * ppy - vby * ppx;
  float ra2 = 1.0f / (sqrtf(fmaf(a2x, a2x, fmaf(a2y, a2y, a2z * a2z))) + 1e-8f);
  a2x *= ra2; a2y *= ra2; a2z *= ra2;

  // RTF8 = F0 @ rotAxis(alpha7, ax1) @ rotAxis(alpha8, ax2)
  float F8[16];
  rotAxis(M2, q3.z, q3.w, a1x, a1y, a1z);
  mm44(T, F0, M2);
  rotAxis(M2, q4.x, q4.y, a2x, a2y, a2z);
  mm44(F8, T, M2);
  store_mat(fbase + 8 * 16, F8);

  // RTF4 = F8 @ RT[3] @ rotX(alpha3) @ rotZ(alpha9)
  float F4[16];
  mm44(F4, F8, RTl + 3 * 16);
  csn(q1.z, q1.w, 1e-6f, c, sn); postRotX(F4, c, sn);
  csn(q4.z, q4.w, 1e-6f, c, sn); postRotZ(F4, c, sn);
  store_mat(fbase + 4 * 16, F4);

  // RTF5..RTF7 chain
  mm44(T, F4, RTl + 4 * 16); csn(q2.x, q2.y, 1e-6f, c, sn); postRotX(T, c, sn);
  store_mat(fbase + 5 * 16, T);
  mm44(F4, T, RTl + 5 * 16); csn(q2.z, q2.w, 1e-6f, c, sn); postRotX(F4, c, sn);
  store_mat(fbase + 6 * 16, F4);
  mm44(T, F4, RTl + 6 * 16); csn(q3.x, q3.y, 1e-6f, c, sn); postRotX(T, c, sn);
  store_mat(fbase + 7 * 16, T);

  // ---------------- atom transforms: gather frame, 4x4 @ vec4 --------------
  // Read frames back from our own just-written stores (same-wave store->load
  // to the same address is hardware-ordered; hits WGP$/L2). Only rows 0..2
  // are needed since we emit xyz only.
  float* ox = out_xyz + (size_t)l * (kNATOMS * 3);
  for (int a = 0; a < kNATOMS; ++a) {
    int fi = BIl[a];
    const float4* fr = reinterpret_cast<const float4*>(fbase + fi * 16);
    float4 r0 = fr[0], r1 = fr[1], r2 = fr[2];
    float4 b = reinterpret_cast<const float4*>(BXl + a * 4)[0];
    ox[a * 3 + 0] = fmaf(r0.x, b.x, fmaf(r0.y, b.y, fmaf(r0.z, b.z, r0.w * b.w)));
    ox[a * 3 + 1] = fmaf(r1.x, b.x, fmaf(r1.y, b.y, fmaf(r1.z, b.z, r1.w * b.w)));
    ox[a * 3 + 2] = fmaf(r2.x, b.x, fmaf(r2.y, b.y, fmaf(r2.z, b.z, r2.w * b.w)));
  }
}

extern "C" void kernel_launch(void* const* d_in, const int* in_sizes, int n_in,
                              void* d_out, int out_size, void* d_ws, size_t ws_size,
                              hipStream_t stream) {
  const float* xyz    = (const float*)d_in[0];
  const float* alphas = (const float*)d_in[1];
  const float* gRT    = (const float*)d_in[2];
  const float* gBX    = (const float*)d_in[3];
  const int*   seq    = (const int*)d_in[4];
  const int*   gBI    = (const int*)d_in[5];

  const int L = in_sizes[4];  // seq element count = B*L
  float* out_frames = (float*)d_out;                       // L*9*16 floats
  float* out_xyz    = (float*)d_out + (size_t)L * 144;     // L*27*3 floats

  dim3 block(256);
  dim3 grid((L + 255) / 256);
  allatom_kernel<<<grid, block, 0, stream>>>(xyz, alphas, gRT, gBX, seq, gBI,
                                             out_frames, out_xyz, L);
  (void)n_in; (void)out_size; (void)d_ws; (void)ws_size;
}